// ModelNew_80908593922644
// MI455X (gfx1250) — compile-verified
//
#include <hip/hip_runtime.h>
#include <hip/hip_bf16.h>

typedef __attribute__((ext_vector_type(2))) float v2f;
typedef __attribute__((ext_vector_type(8))) float v8f;

// ---------------------------------------------------------------------------
// Prep: w_eff[tap][c] = sum_oc weight[c][oc][tap]  (A-matrix layout, 128x32,
// taps 125..127 zero-padded), plus ws[4096] = sum(bias).
// weight: (32, 64, 5,5,5)  -> c stride 8000, oc stride 125.
// ---------------------------------------------------------------------------
__global__ void tconv_prep_kernel(const float* __restrict__ weight,
                                  const float* __restrict__ bias,
                                  float* __restrict__ ws) {
    int idx = blockIdx.x * 256 + threadIdx.x;
    if (idx < 4000) {
        int tap = idx >> 5;       // 0..124
        int c   = idx & 31;       // 0..31
        const float* wp = weight + (size_t)c * 8000 + tap;
        float s = 0.f;
#pragma unroll 8
        for (int oc = 0; oc < 64; ++oc) s += wp[oc * 125];
        ws[idx] = s;
    } else if (idx < 4096) {
        ws[idx] = 0.f;            // zero-pad taps 125..127
    }
    if (idx == 4096) {
        float s = 0.f;
        for (int i = 0; i < 64; ++i) s += bias[i];
        ws[4096] = s;
    }
}

// ---------------------------------------------------------------------------
// Fused: Z = W(128x32) x X(32x32) GEMM per (i_d,i_h) via v_wmma_f32_16x16x4_f32,
// scatter Z[tap,i_w] -> y[2i-2+r] into LDS (ds_add_f32), 6x6x6 maxpool, +bias.
// One workgroup (4 waves) per (n, fd, fh); 16*5*10 = 800 workgroups.
// x: (16,32,16,32,32), strides n:524288 c:16384 d:1024 h:32 w:1.
// ---------------------------------------------------------------------------
__global__ __launch_bounds__(128) void tconv_pool_kernel(
        const float* __restrict__ x,
        const float* __restrict__ ws,    // [0..4095] W A-layout, [4096] bias_sum
        float* __restrict__ out) {
    __shared__ float w_lds[128 * 32];    // A[tap][c]
    __shared__ float y_lds[6 * 6 * 64];  // y slab, o_w padded to 64

    const int tid  = threadIdx.x;
    const int blk  = blockIdx.x;
    const int n    = blk / 50;
    const int rem  = blk % 50;
    const int fd   = rem / 10;           // 0..4
    const int fh   = rem % 10;           // 0..9

    for (int i = tid; i < 6 * 6 * 64; i += 128) y_lds[i] = 0.f;
    for (int i = tid; i < 128 * 32;   i += 128) w_lds[i] = ws[i];
    const float bias_sum = ws[4096];
    __syncthreads();

    const int wave = tid >> 5;
    const int lane = tid & 31;
    const int g    = lane >> 4;          // K-half selector
    const int ln   = lane & 15;          // row/col within tile

    // 25 (i_d, i_h) combos split across 4 waves (wave-uniform control flow).
    for (int combo = wave; combo < 25; combo += 4) {
        const int i_d = 3 * fd - 1 + combo / 5;
        const int i_h = 3 * fh - 1 + combo % 5;
        if (i_d < 0 || i_d >= 16 || i_h >= 32) continue;  // i_h >= -1 always
        if (i_h < 0) continue;

        const float* xb = x + (size_t)n * 524288 + (size_t)i_d * 1024
                            + (size_t)i_h * 32;

        // B fragments: B[c, i_w], lane ln -> column, K = 4*ks + 2*g + j.
        v2f bfrag[8][2];
#pragma unroll
        for (int ks = 0; ks < 8; ++ks) {
#pragma unroll
            for (int nt = 0; nt < 2; ++nt) {
                const int c0 = 4 * ks + 2 * g;
                const int iw = 16 * nt + ln;
                bfrag[ks][nt].x = xb[(size_t)c0 * 16384 + iw];
                bfrag[ks][nt].y = xb[(size_t)(c0 + 1) * 16384 + iw];
            }
        }

#pragma unroll
        for (int nt = 0; nt < 2; ++nt) {
#pragma unroll
            for (int mt = 0; mt < 8; ++mt) {
                v8f acc = {};
                const int arow = 16 * mt + ln;  // A row (tap), K-half via g
#pragma unroll
                for (int ks = 0; ks < 8; ++ks) {
                    v2f a;
                    a.x = w_lds[arow * 32 + 4 * ks + 2 * g];
                    a.y = w_lds[arow * 32 + 4 * ks + 2 * g + 1];
                    acc = __builtin_amdgcn_wmma_f32_16x16x4_f32(
                        /*neg_a=*/false, a, /*neg_b=*/false, bfrag[ks][nt],
                        /*c_mod=*/(short)0, acc,
                        /*reuse_a=*/false, /*reuse_b=*/false);
                }
                // Scatter: D element j -> row (j + 8*g), col ln.
                const int ow_base = 2 * (16 * nt + ln) - 2;
#pragma unroll
                for (int j = 0; j < 8; ++j) {
                    const int tap = 16 * mt + j + 8 * g;
                    if (tap < 125) {
                        const int rd = tap / 25;
                        const int rh = (tap / 5) % 5;
                        const int rw = tap % 5;
                        const int od = 2 * i_d - 2 + rd - 6 * fd;
                        const int oh = 2 * i_h - 2 + rh - 6 * fh;
                        const int ow = ow_base + rw;
                        if (od >= 0 && od < 6 && oh >= 0 && oh < 6 &&
                            ow >= 0 && ow < 60) {
                            atomicAdd(&y_lds[(od * 6 + oh) * 64 + ow], acc[j]);
                        }
                    }
                }
            }
        }
    }

    __syncthreads();

    // 6x6x6 non-overlapping maxpool (= pool2 then pool3) + summed bias.
    if (tid < 10) {
        const int fw = tid;
        float m = -3.402823466e+38f;
        for (int dd = 0; dd < 6; ++dd)
            for (int hh = 0; hh < 6; ++hh)
#pragma unroll
                for (int ww = 0; ww < 6; ++ww)
                    m = fmaxf(m, y_lds[(dd * 6 + hh) * 64 + 6 * fw + ww]);
        out[((n * 5 + fd) * 10 + fh) * 10 + fw] = m + bias_sum;
    }
}

extern "C" void kernel_launch(void* const* d_in, const int* in_sizes, int n_in,
                              void* d_out, int out_size, void* d_ws, size_t ws_size,
                              hipStream_t stream) {
    const float* x      = (const float*)d_in[0];
    const float* weight = (const float*)d_in[1];
    const float* bias   = (const float*)d_in[2];
    float*       out    = (float*)d_out;
    float*       ws     = (float*)d_ws;

    tconv_prep_kernel<<<17, 256, 0, stream>>>(weight, bias, ws);
    tconv_pool_kernel<<<800, 128, 0, stream>>>(x, ws, out);
}